// tempSignalAwareMVNProfile_78314433675745
// MI455X (gfx1250) — compile-verified
//
#include <hip/hip_runtime.h>
#include <math.h>

// CDNA5 / gfx1250. wave32. Two-kernel pipeline:
//  K1: f32 WMMA skinny GEMM (16x16x4_F32) + nonlinear epilogue -> params in d_ws
//  K2: per-sample 1323-pixel Gaussian log-prob + wave-level softmax, streamed out.

typedef __attribute__((ext_vector_type(2))) float v2f;
typedef __attribute__((ext_vector_type(8))) float v8f;

#define DM 1024
#define NPIX 1323
#define NP_ITERS 42            // ceil(1323/32)
#define K_CHUNKS 256           // DM / 4
#define STAGE_CHUNKS 64        // k-chunks staged in LDS per pass (16 KB)

__device__ __forceinline__ float softplus_f(float x) {
    // numerically stable softplus: max(x,0) + log1p(exp(-|x|))
    return fmaxf(x, 0.0f) + log1pf(__expf(-fabsf(x)));
}

__global__ __launch_bounds__(256)
void mvn_gemm_params_kernel(const float* __restrict__ rep,
                            const float* __restrict__ signal_prob,
                            const float* __restrict__ W_mean,
                            const float* __restrict__ b_mean,
                            const float* __restrict__ W_scale,
                            const float* __restrict__ b_scale,
                            float* __restrict__ params,
                            int Btotal)
{
    __shared__ v2f   bfrag[STAGE_CHUNKS][32];   // 16 KB: B fragments (W^T), shared by all 8 waves
    __shared__ float tile[8][16][16];           // 8 KB: per-wave 16x16 WMMA result

    const int tid  = threadIdx.x;
    const int wave = tid >> 5;
    const int lane = tid & 31;
    const int half = lane >> 4;     // 0 => K pair {0,1}, 1 => K pair {2,3}
    const int sub  = lane & 15;

    const int rowBase = blockIdx.x * 128 + wave * 16;
    const float* __restrict__ arow = rep + (size_t)(rowBase + sub) * DM;

    v8f acc = {};

    for (int kb = 0; kb < K_CHUNKS; kb += STAGE_CHUNKS) {
        // Cooperatively build B fragments for chunks [kb, kb+STAGE_CHUNKS).
        // B is W^T (1024 x 9, zero-padded to 16 cols). Fragment layout mirrors A:
        // lane<16: n=lane, holds K={k0,k0+1}; lane>=16: n=lane-16, K={k0+2,k0+3}.
        for (int e = tid; e < STAGE_CHUNKS * 32; e += 256) {
            int c = e >> 5;
            int l = e & 31;
            int n = l & 15;
            int k = (kb + c) * 4 + 2 * (l >> 4);
            v2f v = {};
            if (n < 3) {
                v.x = W_mean[n * DM + k];
                v.y = W_mean[n * DM + k + 1];
            } else if (n < 9) {
                v.x = W_scale[(n - 3) * DM + k];
                v.y = W_scale[(n - 3) * DM + k + 1];
            }
            bfrag[c][l] = v;
        }
        __syncthreads();

        #pragma unroll 8
        for (int c = 0; c < STAGE_CHUNKS; ++c) {
            int k0 = (kb + c) * 4 + 2 * half;
            v2f a = *reinterpret_cast<const v2f*>(arow + k0);  // 8B aligned contiguous pair
            v2f b = bfrag[c][lane];
            // (neg_a, A, neg_b, B, c_mod, C, reuse_a, reuse_b)
            acc = __builtin_amdgcn_wmma_f32_16x16x4_f32(
                false, a, false, b, (short)0, acc, false, false);
        }
        __syncthreads();
    }

    // C/D layout: lane holds column n=sub; VGPR i holds row m = i + 8*half.
    #pragma unroll
    for (int i = 0; i < 8; ++i)
        tile[wave][i + 8 * half][sub] = acc[i];
    __syncthreads();

    // Epilogue: one thread per sample (128 samples per block).
    if (tid < 128) {
        const int sample = blockIdx.x * 128 + tid;
        if (sample < Btotal) {
            const float* r = &tile[tid >> 4][tid & 15][0];

            float m0 = r[0] + b_mean[0];
            float m1 = r[1] + b_mean[1];
            float m2 = r[2] + b_mean[2];

            float sp = fminf(fmaxf(signal_prob[sample], 0.0f), 1.0f);
            float alpha = sp * sp * sp;
            float oma = 1.0f - alpha;

            const float iso[6] = {5.0f, 0.0f, 5.0f, 0.0f, 0.0f, 5.0f};
            float b[6];
            #pragma unroll
            for (int j = 0; j < 6; ++j) {
                float raw = r[3 + j] + b_scale[j];
                float scl = (raw > 0.0f) ? (raw + 1.0f) : __expf(raw);
                b[j] = alpha * scl + oma * iso[j];
            }

            float l00 = softplus_f(b[0]);
            float l11 = softplus_f(b[2]);
            float l22 = softplus_f(b[5]);
            float logdet = __logf(l00) + __logf(l11) + __logf(l22);

            float* __restrict__ P = params + (size_t)sample * 10;
            P[0] = m0;  P[1] = m1;  P[2] = m2;
            P[3] = 1.0f / l00;
            P[4] = b[1];            // l10
            P[5] = 1.0f / l11;
            P[6] = b[3];            // l20
            P[7] = b[4];            // l21
            P[8] = 1.0f / l22;
            P[9] = logdet;
        }
    }
}

__global__ __launch_bounds__(256)
void mvn_softmax_kernel(const float* __restrict__ params,
                        const float* __restrict__ pixpos,
                        float* __restrict__ out,
                        int Btotal)
{
    __shared__ float pix[NPIX * 3];   // 15.9 KB, reused by 8 samples per block
    for (int i = threadIdx.x; i < NPIX * 3; i += 256) pix[i] = pixpos[i];
    __syncthreads();

    const int wave = threadIdx.x >> 5;
    const int lane = threadIdx.x & 31;
    const int sample = blockIdx.x * 8 + wave;
    if (sample >= Btotal) return;

    const float* __restrict__ P = params + (size_t)sample * 10;
    const float m0 = P[0], m1 = P[1], m2 = P[2];
    const float i00 = P[3], l10 = P[4], i11 = P[5];
    const float l20 = P[6], l21 = P[7], i22 = P[8];
    const float cterm = P[9] + 2.7568156420f;   // logdet + 1.5*log(2*pi)

    float lp[NP_ITERS];
    float mx = -INFINITY;

    #pragma unroll
    for (int i = 0; i < NP_ITERS; ++i) {
        int p = i * 32 + lane;
        bool valid = (p < NPIX);
        int pc = valid ? p : 0;
        float d0 = pix[pc * 3 + 0] - m0;
        float d1 = pix[pc * 3 + 1] - m1;
        float d2 = pix[pc * 3 + 2] - m2;
        float z0 = d0 * i00;
        float z1 = (d1 - l10 * z0) * i11;
        float z2 = (d2 - l20 * z0 - l21 * z1) * i22;
        float v = -0.5f * (z0 * z0 + z1 * z1 + z2 * z2) - cterm;
        v = valid ? v : -INFINITY;
        lp[i] = v;
        mx = fmaxf(mx, v);
    }

    // wave32 max reduction
    #pragma unroll
    for (int off = 16; off > 0; off >>= 1)
        mx = fmaxf(mx, __shfl_xor(mx, off, 32));

    float sum = 0.0f;
    #pragma unroll
    for (int i = 0; i < NP_ITERS; ++i) {
        float e = __expf(lp[i] - mx);
        lp[i] = e;
        sum += e;
    }
    #pragma unroll
    for (int off = 16; off > 0; off >>= 1)
        sum += __shfl_xor(sum, off, 32);

    const float inv = 1.0f / (sum + 1e-10f);
    float* __restrict__ orow = out + (size_t)sample * NPIX;
    #pragma unroll
    for (int i = 0; i < NP_ITERS; ++i) {
        int p = i * 32 + lane;
        if (p < NPIX) orow[p] = lp[i] * inv;   // fully coalesced 128B stores per wave
    }
}

extern "C" void kernel_launch(void* const* d_in, const int* in_sizes, int n_in,
                              void* d_out, int out_size, void* d_ws, size_t ws_size,
                              hipStream_t stream) {
    const float* rep = (const float*)d_in[0];   // (B, 1024)
    const float* sig = (const float*)d_in[1];   // (B,)
    const float* Wm  = (const float*)d_in[2];   // (3, 1024)
    const float* bm  = (const float*)d_in[3];   // (3,)
    const float* Ws  = (const float*)d_in[4];   // (6, 1024)
    const float* bs  = (const float*)d_in[5];   // (6,)
    const float* pp  = (const float*)d_in[6];   // (1323, 3)

    const int B = in_sizes[0] / DM;             // 32768
    float* params = (float*)d_ws;               // B * 10 floats

    const int gemmBlocks = (B + 127) / 128;
    mvn_gemm_params_kernel<<<gemmBlocks, 256, 0, stream>>>(rep, sig, Wm, bm, Ws, bs, params, B);

    const int smBlocks = (B + 7) / 8;
    mvn_softmax_kernel<<<smBlocks, 256, 0, stream>>>(params, pp, (float*)d_out, B);
}